// GATModule_23098334117973
// MI455X (gfx1250) — compile-verified
//
#include <hip/hip_runtime.h>
#include <math.h>

#define F_IN   128   // also H*U
#define HEADS  4
#define UNITS  32
#define LPAD   132   // padded LDS row stride (floats) to spread banks

typedef __attribute__((ext_vector_type(2))) float v2f;
typedef __attribute__((ext_vector_type(4))) float v4f;
typedef __attribute__((ext_vector_type(8))) float v8f;

__device__ __forceinline__ float leaky(float x, float s) { return x > 0.0f ? x : s * x; }

__device__ __forceinline__ void atomicMaxF(float* addr, float val) {
  int* ai = reinterpret_cast<int*>(addr);
  int cur = __float_as_int(*addr);
  while (__int_as_float(cur) < val) {
    int prev = atomicCAS(ai, cur, __float_as_int(val));
    if (prev == cur) break;
    cur = prev;
  }
}

// ---- K1: zero / init scratch ------------------------------------------------
__global__ void k_init(float* __restrict__ rst, float* __restrict__ esum,
                       float* __restrict__ emax, float* __restrict__ gsum,
                       float* __restrict__ gcnt, int N, int G) {
  int i = blockIdx.x * blockDim.x + threadIdx.x;
  if (i < N * F_IN) rst[i] = 0.0f;
  if (i < N * HEADS) { esum[i] = 0.0f; emax[i] = -__builtin_inff(); }
  if (i < G * F_IN) gsum[i] = 0.0f;
  if (i < G) gcnt[i] = 0.0f;
}

// ---- K2: z = feat @ W^T via fp32 WMMA ---------------------------------------
// Block: 128 threads = 4 waves. Block computes a 16(row) x 64(col) stripe:
// wave w owns the 16x16 tile at column offset w*16. Both A (feat rows) and
// B (W^T stripe) tiles are staged in LDS with coalesced b128 loads; the
// unrolled K loop is ds_load_b64 + ds_load_b64 + v_wmma, branch-free.
// A 16x4 fp32 fragment: lanes 0-15 hold K={k,k+1}, lanes 16-31 K={k+2,k+3};
// B fragment mirrored (lane = output column within tile).
__global__ void __launch_bounds__(128) k_gemm_z(const float* __restrict__ feat,
                                                const float* __restrict__ Wg,
                                                float* __restrict__ z, int N) {
  __shared__ __align__(16) float Bs[64 * LPAD];  // W stripe: [o within stripe][k]
  __shared__ __align__(16) float Af[16 * LPAD];  // feat tile: [row within tile][k]

  const int tid = threadIdx.x;
  const int m0  = blockIdx.x * 16;
  const int ob  = blockIdx.y * 64;

  // stage W stripe (64 x 128 floats) -- coalesced float4 loads
  for (int i = tid * 4; i < 64 * F_IN; i += 128 * 4) {
    int o = i >> 7, k = i & (F_IN - 1);
    v4f w = *(const v4f*)(Wg + (size_t)(ob + o) * F_IN + k);
    *(v4f*)&Bs[o * LPAD + k] = w;
  }
  // stage feat tile (16 x 128 floats) -- row clamped, no divergent branches
  for (int i = tid * 4; i < 16 * F_IN; i += 128 * 4) {
    int r = i >> 7, k = i & (F_IN - 1);
    int row = m0 + r;
    row = row < N ? row : N - 1;
    v4f a4 = *(const v4f*)(feat + (size_t)row * F_IN + k);
    *(v4f*)&Af[r * LPAD + k] = a4;
  }
  __syncthreads();

  const int lane = tid & 31;
  const int wave = tid >> 5;
  const int o0   = wave * 16;          // column tile within stripe
  const int half = lane >> 4;          // 0: K pair {k,k+1}   1: {k+2,k+3}
  const int ln   = lane & 15;

  v8f acc = {};
#pragma unroll
  for (int k = 0; k < F_IN; k += 4) {
    const int ka = k + 2 * half;
    v2f a = *(const v2f*)&Af[ln * LPAD + ka];
    v2f b = *(const v2f*)&Bs[(o0 + ln) * LPAD + ka];
    acc = __builtin_amdgcn_wmma_f32_16x16x4_f32(false, a, false, b,
                                                (short)0, acc, false, false);
  }

  // C/D layout: VGPR r -> M = r (lanes 0-15) / r+8 (lanes 16-31), N = ln
#pragma unroll
  for (int r = 0; r < 8; ++r) {
    int m = m0 + r + 8 * half;
    if (m < N) z[(size_t)m * F_IN + ob + o0 + ln] = acc[r];
  }
}

// ---- K3: el/er = per-head dot(z, attn) --------------------------------------
__global__ void k_attn(const float* __restrict__ z, const float* __restrict__ al,
                       const float* __restrict__ ar, float* __restrict__ el,
                       float* __restrict__ er, int N) {
  int i = blockIdx.x * blockDim.x + threadIdx.x;   // (n,h)
  if (i >= N * HEADS) return;
  int h = i & (HEADS - 1);
  int n = i >> 2;
  const float* zr = z + (size_t)n * F_IN + h * UNITS;
  float sl = 0.0f, sr = 0.0f;
#pragma unroll
  for (int u = 0; u < UNITS; ++u) {
    float v = zr[u];
    sl += v * al[h * UNITS + u];
    sr += v * ar[h * UNITS + u];
  }
  el[i] = sl;
  er[i] = sr;
}

// ---- K4: edge scores + segment max over dst ---------------------------------
__global__ void k_edge_score(const int* __restrict__ src, const int* __restrict__ dst,
                             const float* __restrict__ el, const float* __restrict__ er,
                             float* __restrict__ eb, float* __restrict__ emax, int E) {
  int i = blockIdx.x * blockDim.x + threadIdx.x;   // (e,h)
  if (i >= E * HEADS) return;
  int h = i & (HEADS - 1);
  int e = i >> 2;
  int s = src[e], d = dst[e];
  float v = leaky(el[s * HEADS + h] + er[d * HEADS + h], 0.2f);
  eb[i] = v;
  atomicMaxF(&emax[d * HEADS + h], v);
}

// ---- K5: exp(e - max) + segment sum ----------------------------------------
__global__ void k_edge_exp(const int* __restrict__ dst, float* __restrict__ eb,
                           const float* __restrict__ emax, float* __restrict__ esum, int E) {
  int i = blockIdx.x * blockDim.x + threadIdx.x;   // (e,h)
  if (i >= E * HEADS) return;
  int h = i & (HEADS - 1);
  int e = i >> 2;
  int d = dst[e];
  float v = expf(eb[i] - emax[d * HEADS + h]);
  eb[i] = v;
  atomicAdd(&esum[d * HEADS + h], v);
}

// ---- K6: rst[dst] += alpha * z[src]  (128 threads per edge, coalesced) ------
__global__ void k_aggregate(const int* __restrict__ src, const int* __restrict__ dst,
                            const float* __restrict__ eb, const float* __restrict__ esum,
                            const float* __restrict__ z, float* __restrict__ rst,
                            int total) {
  int i = blockIdx.x * blockDim.x + threadIdx.x;   // e*128 + c
  if (i >= total) return;
  int c = i & (F_IN - 1);
  int e = i >> 7;
  int h = c >> 5;
  int s = src[e], d = dst[e];
  const float* zsrc = z + (size_t)s * F_IN + c;
  __builtin_prefetch(zsrc, 0, 3);                  // global_prefetch_b8
  float a = eb[e * HEADS + h] / esum[d * HEADS + h];
  atomicAdd(&rst[(size_t)d * F_IN + c], a * (*zsrc));
}

// ---- K7: h = leaky(rst + bias) -> d_out; per-graph sums + counts ------------
__global__ void k_finalize(const float* __restrict__ rst, const float* __restrict__ bias,
                           const int* __restrict__ gid, float* __restrict__ out_h,
                           float* __restrict__ gsum, float* __restrict__ gcnt, int N) {
  int i = blockIdx.x * blockDim.x + threadIdx.x;   // n*128 + c
  if (i >= N * F_IN) return;
  int c = i & (F_IN - 1);
  int n = i >> 7;
  float v = leaky(rst[i] + bias[c], 0.01f);
  out_h[i] = v;
  int g = gid[n];
  atomicAdd(&gsum[(size_t)g * F_IN + c], v);
  if (c == 0) atomicAdd(&gcnt[g], 1.0f);
}

// ---- K8: hg = leaky(mean @ W_fc^T + b_fc) -----------------------------------
__global__ void k_graph_fc(const float* __restrict__ gsum, const float* __restrict__ gcnt,
                           const float* __restrict__ Wfc, const float* __restrict__ bfc,
                           float* __restrict__ out_g, int G) {
  int i = blockIdx.x * blockDim.x + threadIdx.x;   // g*32 + o
  if (i >= G * UNITS) return;
  int o = i & (UNITS - 1);
  int g = i >> 5;
  float inv = 1.0f / fmaxf(gcnt[g], 1.0f);
  float acc = bfc[o];
#pragma unroll 4
  for (int c = 0; c < F_IN; ++c)
    acc += (gsum[(size_t)g * F_IN + c] * inv) * Wfc[o * F_IN + c];
  out_g[i] = leaky(acc, 0.01f);
}

extern "C" void kernel_launch(void* const* d_in, const int* in_sizes, int n_in,
                              void* d_out, int out_size, void* d_ws, size_t ws_size,
                              hipStream_t stream) {
  const float* feat   = (const float*)d_in[0];
  const float* W_gat  = (const float*)d_in[1];
  const float* attn_l = (const float*)d_in[2];
  const float* attn_r = (const float*)d_in[3];
  const float* bias_g = (const float*)d_in[4];
  const float* W_fc   = (const float*)d_in[5];
  const float* b_fc   = (const float*)d_in[6];
  const int*   src    = (const int*)d_in[7];
  const int*   dst    = (const int*)d_in[8];
  const int*   gid    = (const int*)d_in[9];

  const int N = in_sizes[0] / F_IN;
  const int E = in_sizes[7];
  const int G = (out_size - N * F_IN) / UNITS;

  // workspace carve-up (floats)
  float* ws   = (float*)d_ws;
  float* z    = ws;                       // N*128
  float* el   = z    + (size_t)N * F_IN;  // N*4
  float* er   = el   + (size_t)N * HEADS; // N*4
  float* emax = er   + (size_t)N * HEADS; // N*4
  float* esum = emax + (size_t)N * HEADS; // N*4
  float* eb   = esum + (size_t)N * HEADS; // E*4
  float* rst  = eb   + (size_t)E * HEADS; // N*128
  float* gsum = rst  + (size_t)N * F_IN;  // G*128
  float* gcnt = gsum + (size_t)G * F_IN;  // G

  float* out_h = (float*)d_out;            // [N, 128]
  float* out_g = out_h + (size_t)N * F_IN; // [G, 32]

  const int TB = 256;

  // 1) init scratch
  {
    int n = N * F_IN;
    k_init<<<(n + TB - 1) / TB, TB, 0, stream>>>(rst, esum, emax, gsum, gcnt, N, G);
  }
  // 2) z = feat @ W_gat^T  (WMMA fp32; 4 waves/block, 16x64 stripe per block)
  {
    dim3 grid((N + 15) / 16, F_IN / 64);
    k_gemm_z<<<grid, 128, 0, stream>>>(feat, W_gat, z, N);
  }
  // 3) el/er
  {
    int n = N * HEADS;
    k_attn<<<(n + TB - 1) / TB, TB, 0, stream>>>(z, attn_l, attn_r, el, er, N);
  }
  // 4) edge scores + segment max
  {
    int n = E * HEADS;
    k_edge_score<<<(n + TB - 1) / TB, TB, 0, stream>>>(src, dst, el, er, eb, emax, E);
  }
  // 5) exp + segment sum
  {
    int n = E * HEADS;
    k_edge_exp<<<(n + TB - 1) / TB, TB, 0, stream>>>(dst, eb, emax, esum, E);
  }
  // 6) weighted aggregation
  {
    int n = E * F_IN;   // 102.4M
    k_aggregate<<<(n + TB - 1) / TB, TB, 0, stream>>>(src, dst, eb, esum, z, rst, n);
  }
  // 7) node output + graph accumulation
  {
    int n = N * F_IN;
    k_finalize<<<(n + TB - 1) / TB, TB, 0, stream>>>(rst, bias_g, gid, out_h, gsum, gcnt, N);
  }
  // 8) graph readout FC
  {
    int n = G * UNITS;
    k_graph_fc<<<(n + TB - 1) / TB, TB, 0, stream>>>(gsum, gcnt, W_fc, b_fc, out_g, G);
  }
}